// EnhancedGNNModel_50457275793791
// MI455X (gfx1250) — compile-verified
//
#include <hip/hip_runtime.h>
#include <hip/hip_bf16.h>

// ---------------------------------------------------------------------------
// GraphSAGE x2 layers on 3 graphs, MI455X (gfx1250, wave32).
//   agg = scatter-mean(x[src] -> dst); h = relu(agg@Wl1 + bl1 + x@Wr1)
//   aggh = scatter-mean(h);  out_g = aggh@Wl2 + bl2 + h@Wr2
//   d_out = out_0 + 0.5*out_1 + 0.5*out_2
// Matrix path: V_WMMA_F32_16X16X4_F32 (fp32 WMMA, K chained over 128).
// Degree is edge-structure-only -> computed once per graph, reused by layer 2.
// ---------------------------------------------------------------------------

typedef __attribute__((ext_vector_type(2))) float v2f;
typedef __attribute__((ext_vector_type(8))) float v8f;

#define NNODES 100000
#define NEDGES 800000
#define DIM    128

__global__ void k_zero(float* __restrict__ p, int n) {
    int i = blockIdx.x * blockDim.x + threadIdx.x;
    if (i < n) p[i] = 0.0f;
}

// One wave32 per edge: each lane moves 4 floats (float4 gather + 4 f32 atomics).
// with_deg!=0: lane 0 also bumps the destination degree (layer 1 only).
__global__ __launch_bounds__(256)
void k_scatter(const float* __restrict__ feat, const int* __restrict__ src,
               const int* __restrict__ dst, float* __restrict__ accum,
               float* __restrict__ deg, int nE, int with_deg) {
    int gid  = blockIdx.x * blockDim.x + threadIdx.x;
    int wid  = gid >> 5;
    int lane = gid & 31;
    if (wid >= nE) return;
    int s = src[wid];
    int d = dst[wid];
    const float4 v = *(const float4*)(feat + (size_t)s * DIM + lane * 4);
    float* dp = accum + (size_t)d * DIM + lane * 4;
    atomicAdd(dp + 0, v.x);
    atomicAdd(dp + 1, v.y);
    atomicAdd(dp + 2, v.z);
    atomicAdd(dp + 3, v.w);
    if (with_deg && lane == 0) atomicAdd(deg + d, 1.0f);
}

// Workgroup = 8 wave32 = 16-row M strip. Wave w computes the 16x16 tile at
// cols [16w, 16w+16). A-fragments (agg*1/deg and x) staged once in LDS.
// relu_mode==1: out = relu(res) (layer 1, writes h)
// relu_mode==0: out += scale*res (layer 2, accumulates combined output)
__global__ __launch_bounds__(256)
void k_sage_gemm(const float* __restrict__ Aagg, const float* __restrict__ deg,
                 const float* __restrict__ Ax,
                 const float* __restrict__ Wl, const float* __restrict__ bl,
                 const float* __restrict__ Wr,
                 float* __restrict__ out, int relu_mode, float scale) {
    __shared__ float sAgg[16 * DIM];
    __shared__ float sX[16 * DIM];
    __shared__ float sInv[16];

    const int t    = threadIdx.x;
    const int base = blockIdx.x * 16;          // first node row of this strip

    if (t < 16) sInv[t] = 1.0f / fmaxf(deg[base + t], 1.0f);
    __syncthreads();

    // Stage A matrices: 2048 floats each, 8 per thread.
    for (int i = t; i < 16 * DIM; i += 256) {
        int r = i >> 7;
        size_t gi = (size_t)(base + r) * DIM + (i & 127);
        sAgg[i] = Aagg[gi] * sInv[r];
        sX[i]   = Ax[gi];
    }
    __syncthreads();

    const int wave = t >> 5;
    const int lane = t & 31;
    const int col  = wave * 16 + (lane & 15);  // output column (N dim)
    const int koff = (lane >> 4) << 1;         // lanes 16-31 hold K+2,K+3
    const int arow = (lane & 15) << 7;         // A row * DIM

    v8f c = {};
#pragma unroll
    for (int k = 0; k < DIM; k += 4) {
        v2f a, b;
        // agg @ Wl
        a.x = sAgg[arow + k + koff];
        a.y = sAgg[arow + k + koff + 1];
        b.x = Wl[(k + koff) * DIM + col];
        b.y = Wl[(k + koff + 1) * DIM + col];
        c = __builtin_amdgcn_wmma_f32_16x16x4_f32(false, a, false, b,
                                                  (short)0, c, false, false);
        // x @ Wr (same accumulator)
        a.x = sX[arow + k + koff];
        a.y = sX[arow + k + koff + 1];
        b.x = Wr[(k + koff) * DIM + col];
        b.y = Wr[(k + koff + 1) * DIM + col];
        c = __builtin_amdgcn_wmma_f32_16x16x4_f32(false, a, false, b,
                                                  (short)0, c, false, false);
    }

    const float bias  = bl[col];
    const int   rbase = base + ((lane >> 4) << 3);  // lanes 16-31 -> rows M+8
#pragma unroll
    for (int r = 0; r < 8; ++r) {
        float v = c[r] + bias;
        size_t oi = (size_t)(rbase + r) * DIM + col;
        if (relu_mode) out[oi] = fmaxf(v, 0.0f);
        else           out[oi] += scale * v;
    }
}

extern "C" void kernel_launch(void* const* d_in, const int* in_sizes, int n_in,
                              void* d_out, int out_size, void* d_ws, size_t ws_size,
                              hipStream_t stream) {
    const int N = NNODES, E = NEDGES, D = DIM;

    const float* x   = (const float*)d_in[0];
    const int*   ei  = (const int*)d_in[1];   // [3, 2, E] int32
    const float* Wl1 = (const float*)d_in[2];
    const float* bl1 = (const float*)d_in[3];
    const float* Wr1 = (const float*)d_in[4];
    const float* Wl2 = (const float*)d_in[5];
    const float* bl2 = (const float*)d_in[6];
    const float* Wr2 = (const float*)d_in[7];
    float* out = (float*)d_out;

    // Workspace: agg [N*D] | h [N*D] | deg [N]  (~103 MB)
    float* agg = (float*)d_ws;
    float* h   = agg + (size_t)N * D;
    float* deg = h + (size_t)N * D;

    const int nFeat = N * D;
    const int zbF = (nFeat + 255) / 256;
    const int zbD = (N + 255) / 256;
    const int scatterBlocks = (E * 32 + 255) / 256;  // one wave per edge
    const int gemmBlocks = N / 16;                   // 100000 % 16 == 0

    k_zero<<<zbF, 256, 0, stream>>>(out, nFeat);

    for (int g = 0; g < 3; ++g) {
        const float* xg  = x + (size_t)g * N * D;
        const int*   src = ei + (size_t)g * 2 * E;
        const int*   dst = src + E;
        const float* wl1 = Wl1 + (size_t)g * D * D;
        const float* b1  = bl1 + (size_t)g * D;
        const float* wr1 = Wr1 + (size_t)g * D * D;
        const float* wl2 = Wl2 + (size_t)g * D * D;
        const float* b2  = bl2 + (size_t)g * D;
        const float* wr2 = Wr2 + (size_t)g * D * D;

        // Layer 1 (also computes per-graph degree, reused by layer 2)
        k_zero<<<zbF, 256, 0, stream>>>(agg, nFeat);
        k_zero<<<zbD, 256, 0, stream>>>(deg, N);
        k_scatter<<<scatterBlocks, 256, 0, stream>>>(xg, src, dst, agg, deg, E, 1);
        k_sage_gemm<<<gemmBlocks, 256, 0, stream>>>(agg, deg, xg, wl1, b1, wr1,
                                                    h, 1, 1.0f);
        // Layer 2 (degree already known; accumulate combined output)
        k_zero<<<zbF, 256, 0, stream>>>(agg, nFeat);
        k_scatter<<<scatterBlocks, 256, 0, stream>>>(h, src, dst, agg, deg, E, 0);
        k_sage_gemm<<<gemmBlocks, 256, 0, stream>>>(agg, deg, h, wl2, b2, wr2,
                                                    out, 0, (g == 0) ? 1.0f : 0.5f);
    }
}